// NTN_5583457485033
// MI455X (gfx1250) — compile-verified
//
#include <hip/hip_runtime.h>

typedef __attribute__((ext_vector_type(2))) float v2f;
typedef __attribute__((ext_vector_type(8))) float v8f;

#define FM_EPS 1e-12f
#define N_ROWS 32768
#define DDIM   512
#define FDIM   64
#define PITCH  132   // 128 + 4 pad: LDS row stride mod 64 banks = 4 -> conflict free

// ---------------------------------------------------------------------------
// Kernel 1: M[f*512+d] = V[f*1024+d] + sum_e W[(f*512+d)*512+e] * x2[e]
// One wave32 per (f,d) row. float4 coalesced loads, xor-shuffle reduction.
// ---------------------------------------------------------------------------
__global__ __launch_bounds__(256) void fm_precompute_M(
    const float* __restrict__ W, const float* __restrict__ V,
    const float* __restrict__ x2, float* __restrict__ M)
{
    const int lane = threadIdx.x & 31;
    const int wid  = threadIdx.x >> 5;
    const int row  = blockIdx.x * 8 + wid;                 // f*512 + d
    const float* wrow = W + (size_t)row * DDIM;
    float4 p = {0.f, 0.f, 0.f, 0.f};
    #pragma unroll
    for (int i = 0; i < 4; ++i) {
        const int e = i * 128 + lane * 4;
        float4 wv = *(const float4*)(wrow + e);
        float4 xv = *(const float4*)(x2 + e);
        p.x = fmaf(wv.x, xv.x, p.x);
        p.y = fmaf(wv.y, xv.y, p.y);
        p.z = fmaf(wv.z, xv.z, p.z);
        p.w = fmaf(wv.w, xv.w, p.w);
    }
    float s = (p.x + p.y) + (p.z + p.w);
    #pragma unroll
    for (int m = 16; m >= 1; m >>= 1) s += __shfl_xor(s, m, 32);
    if (lane == 0) {
        const int f = row >> 9, d = row & 511;
        M[row] = V[f * 1024 + d] + s;
    }
}

// ---------------------------------------------------------------------------
// Kernel 2: c[f] = b[f] + sum_d V[f*1024 + 512 + d] * x2[d].  One wave per f.
// ---------------------------------------------------------------------------
__global__ __launch_bounds__(256) void fm_precompute_c(
    const float* __restrict__ V, const float* __restrict__ x2,
    const float* __restrict__ b, float* __restrict__ c)
{
    const int lane = threadIdx.x & 31;
    const int wid  = threadIdx.x >> 5;
    const int f    = blockIdx.x * 8 + wid;                 // 0..63
    const float* vrow = V + f * 1024 + DDIM;
    float4 p = {0.f, 0.f, 0.f, 0.f};
    #pragma unroll
    for (int i = 0; i < 4; ++i) {
        const int e = i * 128 + lane * 4;
        float4 vv = *(const float4*)(vrow + e);
        float4 xv = *(const float4*)(x2 + e);
        p.x = fmaf(vv.x, xv.x, p.x);
        p.y = fmaf(vv.y, xv.y, p.y);
        p.z = fmaf(vv.z, xv.z, p.z);
        p.w = fmaf(vv.w, xv.w, p.w);
    }
    float s = (p.x + p.y) + (p.z + p.w);
    #pragma unroll
    for (int m = 16; m >= 1; m >>= 1) s += __shfl_xor(s, m, 32);
    if (lane == 0) c[f] = b[f] + s;
}

// ---------------------------------------------------------------------------
// Kernel 3: fm = x1 @ M^T + c ; out[n] = sum_f relu(fm)/max(sum_f|fm|,eps)*U[f]
// 8 waves/WG; wave w -> rows n0..n0+15, all 64 features via 4 WMMA tiles.
// M chunk (64 x 128 f32) staged in LDS, K-loop of v_wmma_f32_16x16x4_f32.
// ---------------------------------------------------------------------------
__global__ __launch_bounds__(256) void fm_main(
    const float* __restrict__ x1, const float* __restrict__ M,
    const float* __restrict__ c,  const float* __restrict__ U,
    float* __restrict__ out)
{
    __shared__ float lm[FDIM * PITCH];                     // 33 KB

    const int tid  = threadIdx.x;
    const int lane = tid & 31;
    const int wid  = tid >> 5;                             // 0..7
    const int half = lane >> 4;                            // 0 or 1
    const int l16  = lane & 15;
    const int n0   = blockIdx.x * 128 + wid * 16;

    // per-lane feature constants: feature index for tile t is t*16 + l16
    float cv[4], uv[4];
    #pragma unroll
    for (int t = 0; t < 4; ++t) {
        cv[t] = c[t * 16 + l16];
        uv[t] = U[t * 16 + l16];
    }

    v8f acc[4];
    #pragma unroll
    for (int t = 0; t < 4; ++t) acc[t] = {};

    // A fragment base: lane l holds x1[n0 + l%16][k0 + 2*(l/16) + j], j=0,1
    const float* aptr = x1 + (size_t)(n0 + l16) * DDIM + 2 * half;

    for (int ch = 0; ch < 4; ++ch) {
        const int dbase = ch * 128;
        __syncthreads();                                   // chunk consumed
        // cooperative stage M[:, dbase..dbase+127] -> LDS (pitch 132)
        #pragma unroll
        for (int i = 0; i < 8; ++i) {
            const int idx = tid + i * 256;                 // float4 slot 0..2047
            const int r   = idx >> 5;                      // row 0..63
            const int c4  = idx & 31;                      // float4 col 0..31
            float4 v = *(const float4*)(M + r * DDIM + dbase + c4 * 4);
            *(float4*)(&lm[r * PITCH + c4 * 4]) = v;
        }
        __syncthreads();                                   // chunk ready

        #pragma unroll 4
        for (int kk = 0; kk < 32; ++kk) {
            const int d0 = kk * 4;
            v2f a = *(const v2f*)(aptr + dbase + d0);
            #pragma unroll
            for (int t = 0; t < 4; ++t) {
                // B fragment: lane l holds M[t*16 + l%16][d0 + 2*(l/16) + j]
                v2f bf = *(const v2f*)(&lm[(t * 16 + l16) * PITCH + d0 + 2 * half]);
                acc[t] = __builtin_amdgcn_wmma_f32_16x16x4_f32(
                    false, a, false, bf, (short)0, acc[t], false, false);
            }
        }
    }

    // fm = acc + c[f]
    #pragma unroll
    for (int t = 0; t < 4; ++t) {
        #pragma unroll
        for (int r = 0; r < 8; ++r) acc[t][r] += cv[t];
    }

    // D layout: VGPR r, lane l -> fm[n0 + r + 8*(l/16)][f = 16*t + l%16]
    float s[8];
    #pragma unroll
    for (int r = 0; r < 8; ++r) {
        float v = fabsf(acc[0][r]) + fabsf(acc[1][r]) +
                  fabsf(acc[2][r]) + fabsf(acc[3][r]);
        v += __shfl_xor(v, 1, 32);
        v += __shfl_xor(v, 2, 32);
        v += __shfl_xor(v, 4, 32);
        v += __shfl_xor(v, 8, 32);                          // sum over 16 lanes of half
        s[r] = fmaxf(v, FM_EPS);
    }

    #pragma unroll
    for (int r = 0; r < 8; ++r) {
        float p = fmaxf(acc[0][r], 0.f) * uv[0] + fmaxf(acc[1][r], 0.f) * uv[1] +
                  fmaxf(acc[2][r], 0.f) * uv[2] + fmaxf(acc[3][r], 0.f) * uv[3];
        p /= s[r];
        p += __shfl_xor(p, 1, 32);
        p += __shfl_xor(p, 2, 32);
        p += __shfl_xor(p, 4, 32);
        p += __shfl_xor(p, 8, 32);
        if (l16 == 0) out[n0 + half * 8 + r] = p;
    }
}

// ---------------------------------------------------------------------------
extern "C" void kernel_launch(void* const* d_in, const int* in_sizes, int n_in,
                              void* d_out, int out_size, void* d_ws, size_t ws_size,
                              hipStream_t stream)
{
    const float* x1 = (const float*)d_in[0];   // (32768, 512)
    const float* x2 = (const float*)d_in[1];   // (1, 512)
    const float* V  = (const float*)d_in[2];   // (64, 1024)
    const float* W  = (const float*)d_in[3];   // (64, 512, 512)
    const float* b  = (const float*)d_in[4];   // (64,)
    const float* U  = (const float*)d_in[5];   // (64, 1)
    float* out = (float*)d_out;                // (32768, 1)

    float* M = (float*)d_ws;                   // 64*512 floats = 128 KB
    float* c = M + FDIM * DDIM;                // 64 floats

    fm_precompute_M<<<dim3(4096), dim3(256), 0, stream>>>(W, V, x2, M);
    fm_precompute_c<<<dim3(8),    dim3(256), 0, stream>>>(V, x2, b, c);
    fm_main        <<<dim3(256),  dim3(256), 0, stream>>>(x1, M, c, U, out);
}